// AttentionRevisedDecoderRNN_39625368273194
// MI455X (gfx1250) — compile-verified
//
#include <hip/hip_runtime.h>
#include <math.h>

// ---------------- problem constants ----------------
#define HIDDEN     8192
#define OUT_N      6
#define SEQ        6
#define KDIM       4114          // INPUT_SIZE + 18
#define XDIM       4120          // INPUT_SIZE + 24  (LSTM input width)
#define NGATES     24            // 4 * OUT_N

// phase-2 tiling: each block = 8 waves, each wave owns TWO 16-col WMMA tiles
#define HCHUNKS    32
#define HCHUNK     256           // HIDDEN / HCHUNKS
#define WAVES_PB   8
#define KCOLS_PW   32            // columns per wave (2 tiles)
#define KCOLS_PB   (WAVES_PB * KCOLS_PW)   // 256
#define KBLOCKS    17            // ceil(KDIM / 256)
#define AROWS      7             // 6 real rows + 1 shared zero row for M-padding

typedef float v2f __attribute__((ext_vector_type(2)));
typedef float v8f __attribute__((ext_vector_type(8)));

// ---------------- workspace layout (float offsets) ----------------
// attn_w  : SEQ*HIDDEN     = 49152
// partial : HCHUNKS*SEQ*KDIM = 789888
// x       : SEQ*XDIM       = 24720
// xw      : SEQ*NGATES     = 144
// hs      : SEQ*OUT_N      = 36
// y       : SEQ*HIDDEN     = 49152     -> total ~3.66 MB
#define WS_ATTNW   0u
#define WS_PARTIAL (WS_ATTNW + (unsigned)(SEQ*HIDDEN))
#define WS_X       (WS_PARTIAL + (unsigned)(HCHUNKS*SEQ*KDIM))
#define WS_XW      (WS_X + (unsigned)(SEQ*XDIM))
#define WS_HS      (WS_XW + (unsigned)(SEQ*NGATES))
#define WS_Y       (WS_HS + (unsigned)(SEQ*OUT_N))

#define NEG_INF    (-3.402823466e38f)

// ---------------- block-wide reductions (wave32) ----------------
__device__ __forceinline__ float block_reduce_max(float v, float* red) {
  for (int off = 16; off > 0; off >>= 1) v = fmaxf(v, __shfl_xor(v, off, 32));
  const int wave = threadIdx.x >> 5;
  if ((threadIdx.x & 31) == 0) red[wave] = v;
  __syncthreads();
  const int nw = blockDim.x >> 5;
  float r = (threadIdx.x < (unsigned)nw) ? red[threadIdx.x] : NEG_INF;
  if (threadIdx.x < 32) {
    for (int off = 16; off > 0; off >>= 1) r = fmaxf(r, __shfl_xor(r, off, 32));
    if (threadIdx.x == 0) red[0] = r;
  }
  __syncthreads();
  float out = red[0];
  __syncthreads();
  return out;
}

__device__ __forceinline__ float block_reduce_sum(float v, float* red) {
  for (int off = 16; off > 0; off >>= 1) v += __shfl_xor(v, off, 32);
  const int wave = threadIdx.x >> 5;
  if ((threadIdx.x & 31) == 0) red[wave] = v;
  __syncthreads();
  const int nw = blockDim.x >> 5;
  float r = (threadIdx.x < (unsigned)nw) ? red[threadIdx.x] : 0.f;
  if (threadIdx.x < 32) {
    for (int off = 16; off > 0; off >>= 1) r += __shfl_xor(r, off, 32);
    if (threadIdx.x == 0) red[0] = r;
  }
  __syncthreads();
  float out = red[0];
  __syncthreads();
  return out;
}

// ---------------- K1: attention logits + softmax -> attn_w (6 x 8192) --------
__global__ __launch_bounds__(1024)
void k1_attn_softmax(const int* __restrict__ inp, const float* __restrict__ hn,
                     const float* __restrict__ cn, const float* __restrict__ W_attn,
                     const float* __restrict__ b_attn, float* __restrict__ attnw) {
  __shared__ float red[32];
  __shared__ float hcs[12];
  __shared__ int   sinp[SEQ];
  if (threadIdx.x < 6)   { hcs[threadIdx.x] = hn[threadIdx.x]; hcs[6 + threadIdx.x] = cn[threadIdx.x]; }
  if (threadIdx.x < SEQ) sinp[threadIdx.x] = inp[threadIdx.x];
  __syncthreads();

  float lg[8][SEQ];
  #pragma unroll
  for (int hh = 0; hh < 8; ++hh) {
    const int h = threadIdx.x + hh * 1024;
    const float* w = W_attn + (size_t)h * 18;
    float wrow[18];
    #pragma unroll
    for (int j = 0; j < 18; ++j) wrow[j] = w[j];
    float base = b_attn[h];
    #pragma unroll
    for (int j = 0; j < 12; ++j) base += hcs[j] * wrow[6 + j];
    #pragma unroll
    for (int s = 0; s < SEQ; ++s) lg[hh][s] = base + wrow[sinp[s]];
  }

  for (int s = 0; s < SEQ; ++s) {
    float mx = NEG_INF;
    #pragma unroll
    for (int hh = 0; hh < 8; ++hh) mx = fmaxf(mx, lg[hh][s]);
    mx = block_reduce_max(mx, red);
    float e[8], sum = 0.f;
    #pragma unroll
    for (int hh = 0; hh < 8; ++hh) { e[hh] = expf(lg[hh][s] - mx); sum += e[hh]; }
    sum = block_reduce_sum(sum, red);
    const float inv = 1.f / sum;
    #pragma unroll
    for (int hh = 0; hh < 8; ++hh)
      attnw[(size_t)s * HIDDEN + threadIdx.x + hh * 1024] = e[hh] * inv;
  }
}

// ---------------- K2: applied partials via V_WMMA_F32_16X16X4_F32 ------------
// partial[hc][s][k] = sum_{h in chunk hc} attn_w[s][h] * enc[h][k]
// A (16x4, rows 6..15 are a shared zero row -> no EXEC divergence) from LDS;
// B (4x16) streamed from enc; 2 k-tiles per wave so each A feeds 2 WMMAs.
__global__ __launch_bounds__(256)
void k2_attn_apply(const float* __restrict__ enc, const float* __restrict__ attnw,
                   float* __restrict__ partial) {
  __shared__ float s_a[AROWS * HCHUNK];   // rows 0..5 = attn_w chunk, row 6 = zeros
  const int hc = blockIdx.y;
  const int h0 = hc * HCHUNK;

  for (int idx = threadIdx.x; idx < AROWS * HCHUNK; idx += 256) {
    const int s = idx >> 8;             // HCHUNK == 256
    s_a[idx] = (s < SEQ) ? attnw[(size_t)s * HIDDEN + h0 + (idx & (HCHUNK - 1))] : 0.f;
  }
  __syncthreads();

  const int wave  = threadIdx.x >> 5;
  const int lane  = threadIdx.x & 31;
  const int k0    = blockIdx.x * KCOLS_PB + wave * KCOLS_PW;
  const int n     = lane & 15;          // B/D column within tile; also A row M
  const int khalf = lane >> 4;          // 0 -> K{0,1}, 1 -> K{2,3}
  const int kp    = 2 * khalf;
  const int arow  = (n < SEQ) ? n : SEQ;            // rows >=6 read the zero row
  const float* abase = s_a + arow * HCHUNK + kp;

  const int kc0 = (k0 + n      < KDIM) ? (k0 + n)      : (KDIM - 1);
  const int kc1 = (k0 + 16 + n < KDIM) ? (k0 + 16 + n) : (KDIM - 1);
  const float* encb0 = enc + (size_t)(h0 + kp) * KDIM + kc0;
  const float* encb1 = enc + (size_t)(h0 + kp) * KDIM + kc1;

  v8f c0 = {0.f,0.f,0.f,0.f,0.f,0.f,0.f,0.f};
  v8f c1 = {0.f,0.f,0.f,0.f,0.f,0.f,0.f,0.f};

  for (int hh = 0; hh < HCHUNK; hh += 4) {
    // A fragment (documented layout): lanes0-15 K{0,1}, lanes16-31 K{2,3}
    v2f a = *(const v2f*)(abase + hh);
    // B fragments: VGPR0 rows {0,2}, VGPR1 rows {1,3} across lane halves
    const float* p0 = encb0 + (size_t)hh * KDIM;
    const float* p1 = encb1 + (size_t)hh * KDIM;
    v2f b0, b1;
    b0.x = p0[0];  b0.y = p0[KDIM];
    b1.x = p1[0];  b1.y = p1[KDIM];
    __builtin_prefetch(p0 + (size_t)8 * KDIM, 0, 0);  // stream-ahead into GL2
    __builtin_prefetch(p1 + (size_t)8 * KDIM, 0, 0);
    c0 = __builtin_amdgcn_wmma_f32_16x16x4_f32(false, a, false, b0, (short)0, c0,
                                               false, false);
    c1 = __builtin_amdgcn_wmma_f32_16x16x4_f32(false, a, false, b1, (short)0, c1,
                                               false, false);
  }

  // D layout: VGPR j, lanes 0-15 -> row M=j. Rows 0..5 are the real output.
  if (khalf == 0) {
    if (k0 + n < KDIM) {
      #pragma unroll
      for (int j = 0; j < SEQ; ++j)
        partial[((size_t)hc * SEQ + j) * KDIM + (k0 + n)] = c0[j];
    }
    if (k0 + 16 + n < KDIM) {
      #pragma unroll
      for (int j = 0; j < SEQ; ++j)
        partial[((size_t)hc * SEQ + j) * KDIM + (k0 + 16 + n)] = c1[j];
    }
  }
}

// ---------------- K3: reduce partials over h-chunks + build x = [onehot|applied]
__global__ __launch_bounds__(256)
void k3_reduce_concat(const float* __restrict__ partial, const int* __restrict__ inp,
                      float* __restrict__ x) {
  const int k = blockIdx.x * 256 + threadIdx.x;
  if (k < KDIM) {
    #pragma unroll
    for (int s = 0; s < SEQ; ++s) {
      float acc = 0.f;
      for (int hc = 0; hc < HCHUNKS; ++hc)
        acc += partial[((size_t)hc * SEQ + s) * KDIM + k];
      x[(size_t)s * XDIM + 6 + k] = acc;
    }
  }
  if (blockIdx.x == 0 && threadIdx.x < SEQ * 6) {
    const int s = threadIdx.x / 6, j = threadIdx.x % 6;
    x[(size_t)s * XDIM + j] = (inp[s] == j) ? 1.f : 0.f;
  }
}

// ---------------- K4: gate pre-activations xw[s][g] = x[s].W_ih[g] + b_ih + b_hh
__global__ __launch_bounds__(256)
void k4_gates(const float* __restrict__ x, const float* __restrict__ W_ih,
              const float* __restrict__ b_ih, const float* __restrict__ b_hh,
              float* __restrict__ xw) {
  const int wave = (int)((blockIdx.x * blockDim.x + threadIdx.x) >> 5);
  const int lane = threadIdx.x & 31;
  if (wave >= SEQ * NGATES) return;      // grid sized exactly: 18*8 = 144 waves
  const int s = wave / NGATES;
  const int g = wave % NGATES;
  const float* xr = x + (size_t)s * XDIM;
  const float* wr = W_ih + (size_t)g * XDIM;
  float acc = 0.f;
  for (int i = lane; i < XDIM; i += 32) acc += xr[i] * wr[i];
  for (int off = 16; off > 0; off >>= 1) acc += __shfl_xor(acc, off, 32);
  if (lane == 0) xw[wave] = acc + b_ih[g] + b_hh[g];
}

// ---------------- K5: sequential LSTM scan (tiny), writes hs + hT/cT ---------
__global__ __launch_bounds__(32)
void k5_scan(const float* __restrict__ hn, const float* __restrict__ cn,
             const float* __restrict__ W_hh, const float* __restrict__ xw,
             float* __restrict__ hs, float* __restrict__ out) {
  __shared__ float h[6], c[6], g[NGATES];
  const int t = threadIdx.x;
  if (t < 6) { h[t] = hn[t]; c[t] = cn[t]; }
  __syncthreads();
  for (int s = 0; s < SEQ; ++s) {
    if (t < NGATES) {
      float acc = xw[s * NGATES + t];
      #pragma unroll
      for (int j = 0; j < 6; ++j) acc += h[j] * W_hh[t * 6 + j];
      g[t] = acc;
    }
    __syncthreads();
    if (t < 6) {
      const float ig = 1.f / (1.f + expf(-g[t]));
      const float fg = 1.f / (1.f + expf(-g[6 + t]));
      const float gg = tanhf(g[12 + t]);
      const float og = 1.f / (1.f + expf(-g[18 + t]));
      const float c2 = fg * c[t] + ig * gg;
      const float h2 = og * tanhf(c2);
      c[t] = c2; h[t] = h2; hs[s * 6 + t] = h2;
    }
    __syncthreads();
  }
  if (t < 6) { out[36 + t] = h[t]; out[42 + t] = c[t]; }
}

// ---------------- K6: y = relu(hs @ W_out11^T + b_out11)  (6 x 8192) ---------
__global__ __launch_bounds__(256)
void k6_out1(const float* __restrict__ hs, const float* __restrict__ W_out11,
             const float* __restrict__ b_out11, float* __restrict__ y) {
  __shared__ float sh[SEQ * 6];
  if (threadIdx.x < SEQ * 6) sh[threadIdx.x] = hs[threadIdx.x];
  __syncthreads();
  const int hid = blockIdx.x * 256 + threadIdx.x;   // 32 blocks * 256 = 8192
  float w[6];
  #pragma unroll
  for (int j = 0; j < 6; ++j) w[j] = W_out11[(size_t)hid * 6 + j];
  const float b = b_out11[hid];
  #pragma unroll
  for (int s = 0; s < SEQ; ++s) {
    float acc = b;
    #pragma unroll
    for (int j = 0; j < 6; ++j) acc += sh[s * 6 + j] * w[j];
    y[(size_t)s * HIDDEN + hid] = fmaxf(acc, 0.f);
  }
}

// ---------------- K7: y2 = relu(y @ W_out12^T + b), then log_softmax ---------
__global__ __launch_bounds__(1024)
void k7_out2_logsoftmax(const float* __restrict__ y, const float* __restrict__ W_out12,
                        const float* __restrict__ b_out12, float* __restrict__ out) {
  __shared__ float y2[36];
  const int wave = threadIdx.x >> 5, lane = threadIdx.x & 31;
  for (int p = wave; p < 36; p += 32) {
    const int s = p / 6, o = p % 6;
    const float* yr = y + (size_t)s * HIDDEN;
    const float* wr = W_out12 + (size_t)o * HIDDEN;
    float acc = 0.f;
    for (int i = lane; i < HIDDEN; i += 32) acc += yr[i] * wr[i];
    for (int off = 16; off > 0; off >>= 1) acc += __shfl_xor(acc, off, 32);
    if (lane == 0) y2[p] = fmaxf(acc + b_out12[o], 0.f);
  }
  __syncthreads();
  if (threadIdx.x < 6) {
    const int s = threadIdx.x;
    float mx = NEG_INF;
    #pragma unroll
    for (int j = 0; j < 6; ++j) mx = fmaxf(mx, y2[s * 6 + j]);
    float sum = 0.f;
    #pragma unroll
    for (int j = 0; j < 6; ++j) sum += expf(y2[s * 6 + j] - mx);
    const float lse = mx + logf(sum);
    #pragma unroll
    for (int j = 0; j < 6; ++j) out[s * 6 + j] = y2[s * 6 + j] - lse;
  }
}

// ---------------- host launcher ----------------
extern "C" void kernel_launch(void* const* d_in, const int* in_sizes, int n_in,
                              void* d_out, int out_size, void* d_ws, size_t ws_size,
                              hipStream_t stream) {
  (void)in_sizes; (void)n_in; (void)out_size; (void)ws_size;
  const int*   inp      = (const int*)  d_in[0];
  const float* hn       = (const float*)d_in[1];
  const float* cn       = (const float*)d_in[2];
  const float* enc      = (const float*)d_in[3];
  const float* W_attn   = (const float*)d_in[4];
  const float* b_attn   = (const float*)d_in[5];
  const float* W_ih     = (const float*)d_in[6];
  const float* W_hh     = (const float*)d_in[7];
  const float* b_ih     = (const float*)d_in[8];
  const float* b_hh     = (const float*)d_in[9];
  const float* W_out11  = (const float*)d_in[10];
  const float* b_out11  = (const float*)d_in[11];
  const float* W_out12  = (const float*)d_in[12];
  const float* b_out12  = (const float*)d_in[13];
  float* out = (float*)d_out;
  float* ws  = (float*)d_ws;

  float* attnw   = ws + WS_ATTNW;
  float* partial = ws + WS_PARTIAL;
  float* x       = ws + WS_X;
  float* xw      = ws + WS_XW;
  float* hs      = ws + WS_HS;
  float* y       = ws + WS_Y;

  k1_attn_softmax<<<1, 1024, 0, stream>>>(inp, hn, cn, W_attn, b_attn, attnw);
  k2_attn_apply<<<dim3(KBLOCKS, HCHUNKS), 256, 0, stream>>>(enc, attnw, partial);
  k3_reduce_concat<<<(KDIM + 255) / 256, 256, 0, stream>>>(partial, inp, x);
  k4_gates<<<(SEQ * NGATES) / 8, 256, 0, stream>>>(x, W_ih, b_ih, b_hh, xw);
  k5_scan<<<1, 32, 0, stream>>>(hn, cn, W_hh, xw, hs, out);
  k6_out1<<<HIDDEN / 256, 256, 0, stream>>>(hs, W_out11, b_out11, y);
  k7_out2_logsoftmax<<<1, 1024, 0, stream>>>(y, W_out12, b_out12, out);
}